// AdditiveAttHead_14972255994661
// MI455X (gfx1250) — compile-verified
//
#include <hip/hip_runtime.h>
#include <hip/hip_bf16.h>

// ---------------------------------------------------------------------------
// GAT additive-attention head, fused single pass over adj (268 MB = the
// roofline floor at 23.3 TB/s).  f32 WMMA (16x16x4) for attention @ h so the
// result matches the f32 reference; softmax stabilized with a global upper
// bound G = max(s_src)+max(s_dst) so adj is read exactly once.
// h-chunk staging uses GLOBAL_LOAD_ASYNC_TO_LDS (ASYNCcnt) when available so
// the h traffic overlaps the mask/exp VALU phase.
// ---------------------------------------------------------------------------

#define GAT_N    8192
#define GAT_EMB  256
#define GAT_NHID 64
#define GAT_ALPHA 0.2f

#define ROWS   32     // output rows per block
#define CHUNK  128    // columns (K of the big matmul) per iteration
#define PT_STRIDE 132 // p_tile row stride (floats): 4*r banks -> conflict-free b64 A loads
#define HT_STRIDE 72  // h_tile row stride (floats): lanes 0-15 / 16-31 hit disjoint banks

typedef float v2f __attribute__((ext_vector_type(2)));
typedef float v8f __attribute__((ext_vector_type(8)));

#if defined(__gfx1250__) && __has_builtin(__builtin_amdgcn_global_load_async_to_lds_b128) && \
    __has_builtin(__builtin_amdgcn_s_wait_asynccnt)
#define USE_ASYNC_COPY 1
// Builtin signature (from hipcc diagnostic): param0 is
// '__attribute__((__vector_size__(4 * sizeof(int)))) int __device__ *'
// i.e. a non-const int4* in the global (addrspace(1)) address space.
typedef int gat_v4i __attribute__((vector_size(4 * sizeof(int))));
typedef __attribute__((address_space(1))) gat_v4i* gat_gptr;
typedef __attribute__((address_space(3))) gat_v4i* gat_lptr;
#else
#define USE_ASYNC_COPY 0
#endif

// ---------------------------------------------------------------------------
// Kernel 1: h = input @ W ; s_src = h @ a[:64] ; s_dst = h @ a[64:]
// 134M MACs total -> plain VALU is fine; one block per row.
// ---------------------------------------------------------------------------
__global__ __launch_bounds__(64) void gat_h_kernel(
    const float* __restrict__ inp, const float* __restrict__ W,
    const float* __restrict__ a, float* __restrict__ h,
    float* __restrict__ s_src, float* __restrict__ s_dst) {
  __shared__ float row[GAT_EMB];
  __shared__ float psrc[GAT_NHID];
  __shared__ float pdst[GAT_NHID];
  const int i = blockIdx.x;
  const int k = threadIdx.x;  // 0..63

  // stage the input row (256 floats, 4 per thread)
  ((float4*)row)[k] = ((const float4*)(inp + (size_t)i * GAT_EMB))[k];
  __syncthreads();

  float acc = 0.f;
#pragma unroll 8
  for (int e = 0; e < GAT_EMB; ++e) acc += row[e] * W[e * GAT_NHID + k];

  h[(size_t)i * GAT_NHID + k] = acc;
  psrc[k] = acc * a[k];
  pdst[k] = acc * a[GAT_NHID + k];
  __syncthreads();

  if (k == 0) {
    float s1 = 0.f, s2 = 0.f;
#pragma unroll
    for (int u = 0; u < GAT_NHID; ++u) { s1 += psrc[u]; s2 += pdst[u]; }
    s_src[i] = s1;
    s_dst[i] = s2;
  }
}

// ---------------------------------------------------------------------------
// Kernel 1b: G = max(s_src) + max(s_dst)  (global softmax shift bound)
// ---------------------------------------------------------------------------
__global__ __launch_bounds__(1024) void gat_gmax_kernel(
    const float* __restrict__ s_src, const float* __restrict__ s_dst,
    float* __restrict__ gmax) {
  __shared__ float m1[1024];
  __shared__ float m2[1024];
  const int t = threadIdx.x;
  float a1 = -3.0e38f, a2 = -3.0e38f;
  for (int i = t; i < GAT_N; i += 1024) {
    a1 = fmaxf(a1, s_src[i]);
    a2 = fmaxf(a2, s_dst[i]);
  }
  m1[t] = a1; m2[t] = a2;
  __syncthreads();
  for (int s = 512; s > 0; s >>= 1) {
    if (t < s) { m1[t] = fmaxf(m1[t], m1[t + s]); m2[t] = fmaxf(m2[t], m2[t + s]); }
    __syncthreads();
  }
  if (t == 0) *gmax = m1[0] + m2[0];
}

// ---------------------------------------------------------------------------
// Kernel 2: fused mask -> exp(e-G) -> (p @ h) -> /rowsum.
// 256 threads = 8 waves; wave w owns the 16x16 output tile (mt = w>>2, nt = w&3)
// of a 32-row x 64-col block, accumulating over all 8192 columns in 128-chunks.
// ---------------------------------------------------------------------------
__global__ __launch_bounds__(256) void gat_attn_kernel(
    const int* __restrict__ adj, const float* __restrict__ h,
    const float* __restrict__ s_src, const float* __restrict__ s_dst,
    const float* __restrict__ gmax_p, float* __restrict__ out) {
  __shared__ float p_tile[ROWS * PT_STRIDE];   // exp weights, padded
  __shared__ float h_tile[CHUNK * HT_STRIDE];  // h chunk, padded
  __shared__ float sdst_sh[CHUNK];
  __shared__ float ssrc_sh[ROWS];
  __shared__ float partial[256];
  __shared__ float lsum_sh[ROWS];

  const int t  = threadIdx.x;
  const int i0 = blockIdx.x * ROWS;
  const float G = *gmax_p;

  if (t < ROWS) ssrc_sh[t] = s_src[i0 + t];
  __syncthreads();

  // step-A role: thread handles row m, 16 consecutive columns in group jg
  const int m  = t >> 3;  // 0..31
  const int jg = t & 7;   // 0..7
  const float  ssrc_m  = ssrc_sh[m];
  const size_t adj_row = (size_t)(i0 + m) * (size_t)GAT_N;

  // step-B role: wave / lane decomposition for WMMA fragments
  const int w    = t >> 5;   // wave 0..7
  const int mt   = w >> 2;   // 0..1 (row tile)
  const int nt   = w & 3;    // 0..3 (col tile)
  const int lane = t & 31;
  const int lm   = lane & 15;
  const int lh   = lane >> 4;  // 0: K=+0/+1 ; 1: K=+2/+3  (ISA 32-bit A/B striping)

  // h-staging role: row jl (0..127), half qh (32 floats each)
  const int jl = t >> 1;
  const int qh = t & 1;

  v8f acc = {0.f, 0.f, 0.f, 0.f, 0.f, 0.f, 0.f, 0.f};
  float lsum = 0.f;

  for (int j0 = 0; j0 < GAT_N; j0 += CHUNK) {
    // ---- stage h chunk into LDS (async if available) -------------------
    {
      const float* src = h + (size_t)(j0 + jl) * GAT_NHID + qh * 32;
      float* dst = &h_tile[jl * HT_STRIDE + qh * 32];
#if USE_ASYNC_COPY
#pragma unroll
      for (int u = 0; u < 8; ++u)
        __builtin_amdgcn_global_load_async_to_lds_b128(
            (gat_gptr)(src + 4 * u), (gat_lptr)(void*)(dst + 4 * u), 0, 0);
#else
#pragma unroll
      for (int u = 0; u < 8; ++u)
        ((float4*)dst)[u] = ((const float4*)src)[u];
#endif
    }
    // ---- stage s_dst chunk ---------------------------------------------
    if (t < CHUNK) sdst_sh[t] = s_dst[j0 + t];
    // prefetch next adj chunk (global_prefetch_b8)
    if (j0 + CHUNK < GAT_N)
      __builtin_prefetch(adj + adj_row + j0 + CHUNK + jg * 16, 0, 1);
    __syncthreads();  // sdst_sh (and fallback h_tile) visible; async may be in flight

    // ---- step A: masked exp weights into p_tile (overlaps async h) -----
    {
      const int4* av = (const int4*)(adj + adj_row + j0 + jg * 16);
      float* prow = &p_tile[m * PT_STRIDE + jg * 16];
#pragma unroll
      for (int q = 0; q < 4; ++q) {
        int4 a4 = av[q];
#pragma unroll
        for (int u = 0; u < 4; ++u) {
          const int c  = q * 4 + u;
          const int ad = (&a4.x)[u];
          const float x  = ssrc_m + sdst_sh[jg * 16 + c];
          const float el = x > 0.f ? x : GAT_ALPHA * x;
          const float pv = (ad > 0) ? __expf(el - G) : 0.f;
          prow[c] = pv;
          lsum += pv;
        }
      }
    }
#if USE_ASYNC_COPY
    __builtin_amdgcn_s_wait_asynccnt(0);  // this wave's h-chunk copies done
#endif
    __syncthreads();  // p_tile + h_tile fully visible to all waves

    // ---- step B: acc += p_tile(16xK) @ h_tile(Kx16) via f32 WMMA -------
    {
      const int arow = (mt * 16 + lm) * PT_STRIDE;
      const int bcol = nt * 16 + lm;
#pragma unroll
      for (int kk = 0; kk < CHUNK / 4; ++kk) {
        const int K0 = kk * 4 + lh * 2;
        v2f av2 = *(const v2f*)&p_tile[arow + K0];     // A: rows, K pair
        v2f bv2;                                       // B: K pair, cols
        bv2.x = h_tile[K0 * HT_STRIDE + bcol];
        bv2.y = h_tile[(K0 + 1) * HT_STRIDE + bcol];
        acc = __builtin_amdgcn_wmma_f32_16x16x4_f32(
            false, av2, false, bv2, (short)0, acc, false, false);
      }
    }
    __syncthreads();  // tiles consumed; safe to overwrite next chunk
  }

  // ---- row denominators (8 partials per row) ---------------------------
  partial[t] = lsum;
  __syncthreads();
  if (t < ROWS) {
    float s = 0.f;
#pragma unroll
    for (int u = 0; u < 8; ++u) s += partial[t * 8 + u];
    lsum_sh[t] = s;
  }
  __syncthreads();

  // ---- normalize + store (C/D layout: VGPR r -> rows r and r+8) --------
#pragma unroll
  for (int r = 0; r < 8; ++r) {
    const int row = mt * 16 + lh * 8 + r;
    const float denom = lsum_sh[row];
    out[(size_t)(i0 + row) * GAT_NHID + nt * 16 + lm] = acc[r] / denom;
  }
}

// ---------------------------------------------------------------------------
extern "C" void kernel_launch(void* const* d_in, const int* in_sizes, int n_in,
                              void* d_out, int out_size, void* d_ws, size_t ws_size,
                              hipStream_t stream) {
  const float* inp = (const float*)d_in[0];
  const int*   adj = (const int*)d_in[1];
  const float* W   = (const float*)d_in[2];
  const float* a   = (const float*)d_in[3];
  float* out = (float*)d_out;

  // workspace layout: h [N*NHID] | s_src [N] | s_dst [N] | gmax [1]
  float* h     = (float*)d_ws;
  float* s_src = h + (size_t)GAT_N * GAT_NHID;
  float* s_dst = s_src + GAT_N;
  float* gmax  = s_dst + GAT_N;

  gat_h_kernel<<<GAT_N, 64, 0, stream>>>(inp, W, a, h, s_src, s_dst);
  gat_gmax_kernel<<<1, 1024, 0, stream>>>(s_src, s_dst, gmax);
  gat_attn_kernel<<<GAT_N / ROWS, 256, 0, stream>>>(adj, h, s_src, s_dst, gmax, out);
}